// Likelyloss_heart_1116691497271
// MI455X (gfx1250) — compile-verified
//
#include <hip/hip_runtime.h>
#include <math.h>

// -------------------------------------------------------------------------
// MI455X (gfx1250) implementation.
// Memory-bound loss reduction: two streaming passes (float4/int4 loads),
// block-level reduction of up to 16 simultaneous f32 statistics done with
// chained V_WMMA_F32_16X16X4_F32 (A = 16x4 LDS tile, B = ones, C accumulate).
// -------------------------------------------------------------------------

typedef __attribute__((ext_vector_type(2))) float v2f;
typedef __attribute__((ext_vector_type(8))) float v8f;

#define BLK 256
#define RP  260                 // LDS row pitch (260 % 64 == 4 -> conflict-free)
#define GRIDX 64                // blocks per batch image (65536/ (64*256) = 4 iters)

// Reduce nst (<=16) per-thread statistics across a 256-thread block using
// wave 0's matrix unit, then atomically add the totals into dst[0..nst).
__device__ __forceinline__ void block_reduce_wmma_atomic(const float* st, int nst,
                                                         float* dst) {
  __shared__ float red[16 * RP];
  const int tid = threadIdx.x;
#pragma unroll
  for (int j = 0; j < 16; ++j)
    red[j * RP + tid] = (j < nst) ? st[j] : 0.0f;
  __syncthreads();

  if (tid < 32) {                       // wave 0, EXEC all ones
    const int m    = tid & 15;          // A-matrix row
    const int koff = (tid >> 4) << 1;   // lanes 0-15: K=0,1 ; lanes 16-31: K=2,3
    const float* row = &red[m * RP + koff];
    v2f bones; bones[0] = 1.0f; bones[1] = 1.0f;   // B = 4x16 all-ones
    v8f c = {};
#pragma unroll 4
    for (int col = 0; col < BLK; col += 4) {
      v2f a;
      a[0] = row[col];
      a[1] = row[col + 1];
      // D[m,n] += sum_k A[m,k] : row sums of the 16x4 tile, accumulated in C
      c = __builtin_amdgcn_wmma_f32_16x16x4_f32(false, a, false, bones,
                                                (short)0, c, false, false);
    }
    // C/D layout: lanes 0-15 VGPR j -> M=j ; lanes 16-31 VGPR j -> M=8+j
    if (tid == 0) {
#pragma unroll
      for (int j = 0; j < 8; ++j)
        if (j < nst) atomicAdd(&dst[j], c[j]);
    } else if (tid == 16) {
#pragma unroll
      for (int j = 0; j < 8; ++j)
        if (j + 8 < nst) atomicAdd(&dst[j + 8], c[j]);
    }
  }
}

__global__ void k_zero(float* ws, int n) {
  const int i = blockIdx.x * blockDim.x + threadIdx.x;
  if (i < n) ws[i] = 0.0f;
}

// Pass 1: per (b,k in l_all order {blood,scar,muscle,mvo}) weighted moments.
// stats[b*16 + {0..3}] = sum(w), {4..7} = sum(w*x), {8..11} = sum(w*x^2),
// stats[b*16 + 12] = sum(m).
__global__ void k_stats(const float* __restrict__ pred,
                        const float* __restrict__ xin,
                        const int*   __restrict__ mask,
                        float* __restrict__ stats) {
  const int b   = blockIdx.y;
  const int HW  = 512 * 512;
  const int HW4 = HW / 4;

  const float4* xp = (const float4*)(xin + (size_t)b * HW);
  const int4*   mp = (const int4*)(mask + (size_t)b * HW);
  const size_t  pb = (size_t)b * 5 * HW;
  const float4* l0 = (const float4*)(pred + pb + (size_t)1 * HW);  // blood
  const float4* l1 = (const float4*)(pred + pb + (size_t)3 * HW);  // scar
  const float4* l2 = (const float4*)(pred + pb + (size_t)2 * HW);  // muscle
  const float4* l3 = (const float4*)(pred + pb + (size_t)4 * HW);  // mvo

  float sw0=0.f,sw1=0.f,sw2=0.f,sw3=0.f;
  float a0=0.f,a1=0.f,a2=0.f,a3=0.f;
  float q0=0.f,q1=0.f,q2=0.f,q3=0.f;
  float cnt=0.f;

  const int stride = GRIDX * BLK;
  for (int i = blockIdx.x * BLK + threadIdx.x; i < HW4; i += stride) {
    const float4 xv = xp[i];
    const int4   mv = mp[i];
    const float4 c0 = l0[i];
    const float4 c1 = l1[i];
    const float4 c2 = l2[i];
    const float4 c3 = l3[i];
    const int nx = i + stride;
    if (nx < HW4) {
      __builtin_prefetch(&xp[nx], 0, 0);
      __builtin_prefetch(&mp[nx], 0, 0);
      __builtin_prefetch(&l0[nx], 0, 0);
      __builtin_prefetch(&l1[nx], 0, 0);
      __builtin_prefetch(&l2[nx], 0, 0);
      __builtin_prefetch(&l3[nx], 0, 0);
    }
#define ACC(comp)                                                         \
    {                                                                     \
      const float mm = (mv.comp == 1) ? 1.0f : 0.0f;                      \
      const float xe = xv.comp;                                           \
      const float x2 = xe * xe;                                           \
      cnt += mm;                                                          \
      float w;                                                            \
      w = c0.comp * mm; sw0 += w; a0 += w * xe; q0 += w * x2;             \
      w = c1.comp * mm; sw1 += w; a1 += w * xe; q1 += w * x2;             \
      w = c2.comp * mm; sw2 += w; a2 += w * xe; q2 += w * x2;             \
      w = c3.comp * mm; sw3 += w; a3 += w * xe; q3 += w * x2;             \
    }
    ACC(x) ACC(y) ACC(z) ACC(w)
#undef ACC
  }

  float st[13] = {sw0, sw1, sw2, sw3, a0, a1, a2, a3, q0, q1, q2, q3, cnt};
  block_reduce_wmma_atomic(st, 13, &stats[b * 16]);
}

// Tiny: stats -> (mu, 1/(2 var), 1/sqrt(2 pi var)) per (b,k). 128 threads.
__global__ void k_params(const float* __restrict__ stats,
                         float* __restrict__ params) {
  const int t = blockIdx.x * blockDim.x + threadIdx.x;
  if (t >= 32 * 4) return;
  const int b = t >> 2, k = t & 3;
  const float EPS = 1.1920928955078125e-7f;
  const float sw = stats[b * 16 + k];
  const float s1 = stats[b * 16 + 4 + k];
  const float s2 = stats[b * 16 + 8 + k];
  const float denom = sw + EPS;
  const float mu  = s1 / denom;
  // sum w (x-mu)^2 == s2 - 2 mu s1 + mu^2 sw
  const float var = (s2 - 2.0f * mu * s1 + mu * mu * sw) / denom + EPS;
  params[b * 12 + k]     = mu;
  params[b * 12 + 4 + k] = 0.5f / var;
  params[b * 12 + 8 + k] = rsqrtf(2.0f * 3.14159265358979323846f * var);
}

// Pass 2: per-batch accumulation of sum(m * log(mix + eps)).
__global__ void k_loss(const float* __restrict__ pred,
                       const float* __restrict__ xin,
                       const int*   __restrict__ mask,
                       const float* __restrict__ params,
                       float* __restrict__ lossacc) {
  const int b   = blockIdx.y;
  const int HW  = 512 * 512;
  const int HW4 = HW / 4;
  const float EPS = 1.1920928955078125e-7f;

  const float* P = &params[b * 12];
  const float mu0 = P[0], mu1 = P[1], mu2 = P[2], mu3 = P[3];
  const float iv0 = P[4], iv1 = P[5], iv2 = P[6], iv3 = P[7];
  const float nr0 = P[8], nr1 = P[9], nr2 = P[10], nr3 = P[11];

  const float4* xp = (const float4*)(xin + (size_t)b * HW);
  const int4*   mp = (const int4*)(mask + (size_t)b * HW);
  const size_t  pb = (size_t)b * 5 * HW;
  const float4* pbld = (const float4*)(pred + pb + (size_t)1 * HW);
  const float4* pmus = (const float4*)(pred + pb + (size_t)2 * HW);
  const float4* pscr = (const float4*)(pred + pb + (size_t)3 * HW);
  const float4* pmvo = (const float4*)(pred + pb + (size_t)4 * HW);

  float acc = 0.0f;
  const int stride = GRIDX * BLK;
  for (int i = blockIdx.x * BLK + threadIdx.x; i < HW4; i += stride) {
    const float4 xv = xp[i];
    const int4   mv = mp[i];
    const float4 bv = pbld[i];
    const float4 uv = pmus[i];
    const float4 sv = pscr[i];
    const float4 ov = pmvo[i];
    const int nx = i + stride;
    if (nx < HW4) {
      __builtin_prefetch(&xp[nx],   0, 0);
      __builtin_prefetch(&mp[nx],   0, 0);
      __builtin_prefetch(&pbld[nx], 0, 0);
      __builtin_prefetch(&pmus[nx], 0, 0);
      __builtin_prefetch(&pscr[nx], 0, 0);
      __builtin_prefetch(&pmvo[nx], 0, 0);
    }
#define LT(comp)                                                          \
    {                                                                     \
      const float mm = (mv.comp == 1) ? 1.0f : 0.0f;                      \
      const float xe = xv.comp;                                           \
      const float bl = bv.comp, mus = uv.comp, sc = sv.comp, mo = ov.comp;\
      const float cls0 = bl + sc, cls1 = mus + mo;                        \
      const float e1 = __expf(sc - bl);                                   \
      const float r1 = 1.0f / (1.0f + e1);                                \
      const float c10 = r1, c11 = e1 * r1;                                \
      const float e2 = __expf(mo - mus);                                  \
      const float r2 = 1.0f / (1.0f + e2);                                \
      const float c20 = r2, c21 = e2 * r2;                                \
      float d;                                                            \
      d = xe - mu0; const float g0 = nr0 * __expf(-d * d * iv0);          \
      d = xe - mu1; const float g1 = nr1 * __expf(-d * d * iv1);          \
      d = xe - mu2; const float g2 = nr2 * __expf(-d * d * iv2);          \
      d = xe - mu3; const float g3 = nr3 * __expf(-d * d * iv3);          \
      const float s0 = c10 * g0 + c11 * g2;                               \
      const float s1 = c20 * g1 + c21 * g3;                               \
      const float mix = cls0 * s0 + cls1 * s1;                            \
      acc += mm * __logf(mix + EPS);                                      \
    }
    LT(x) LT(y) LT(z) LT(w)
#undef LT
  }

  block_reduce_wmma_atomic(&acc, 1, &lossacc[b]);
}

// loss_b = -sum(m*log)/max(cnt,1); out = mean over 32 batches.
__global__ void k_final(const float* __restrict__ stats,
                        const float* __restrict__ lossacc,
                        float* __restrict__ out) {
  __shared__ float sh[32];
  const int t = threadIdx.x;
  float v = 0.0f;
  if (t < 32) {
    const float cnt = stats[t * 16 + 12];
    v = -lossacc[t] / fmaxf(cnt, 1.0f);
  }
  sh[t] = v;
  __syncthreads();
  if (t == 0) {
    float s = 0.0f;
    for (int i = 0; i < 32; ++i) s += sh[i];
    out[0] = s / 32.0f;
  }
}

extern "C" void kernel_launch(void* const* d_in, const int* in_sizes, int n_in,
                              void* d_out, int out_size, void* d_ws, size_t ws_size,
                              hipStream_t stream) {
  (void)in_sizes; (void)n_in; (void)out_size; (void)ws_size;
  const float* pred = (const float*)d_in[0];
  const float* xin  = (const float*)d_in[1];
  const int*   mask = (const int*)d_in[2];
  float* out = (float*)d_out;
  float* ws  = (float*)d_ws;

  float* stats   = ws;        // 32 * 16
  float* params  = ws + 512;  // 32 * 12
  float* lossacc = ws + 896;  // 32
  const int nzero = 928;

  k_zero<<<(nzero + 255) / 256, 256, 0, stream>>>(ws, nzero);

  dim3 grid(GRIDX, 32);
  k_stats<<<grid, BLK, 0, stream>>>(pred, xin, mask, stats);
  k_params<<<1, 128, 0, stream>>>(stats, params);
  k_loss<<<grid, BLK, 0, stream>>>(pred, xin, mask, params, lossacc);
  k_final<<<1, 32, 0, stream>>>(stats, lossacc, out);
}